// GaussianPointMatcher_22763326669100
// MI455X (gfx1250) — compile-verified
//
#include <hip/hip_runtime.h>
#include <math.h>

// ---------------------------------------------------------------------------
// GaussianPointMatcher for MI455X (gfx1250, wave32).
// score[p,g] = Phi(p) . Psi(g)  with K=12  ->  3x V_WMMA_F32_16X16X4_F32 / tile
// ---------------------------------------------------------------------------

typedef __attribute__((ext_vector_type(2))) float v2f;
typedef __attribute__((ext_vector_type(8))) float v8f;

#define LOG_2PI_F 1.8378770664093453f

// ---------------------------------------------------------------------------
// Kernel 1: per-Gaussian setup.  Writes:
//   psi[12][G]      (dim-major, stride G)  : GEMM B-matrix rows
//   minmax[7][G]    : min_c.xyz, max_c.xyz, radius.x   (for quantile sorts)
// ---------------------------------------------------------------------------
__global__ void gauss_setup_kernel(const float* __restrict__ pos,
                                   const float* __restrict__ scl,
                                   const float* __restrict__ quat,
                                   float* __restrict__ psi,
                                   float* __restrict__ minmax,
                                   int G)
{
    int g = blockIdx.x * blockDim.x + threadIdx.x;
    if (g >= G) return;

    float qw = quat[4*g+0], qx = quat[4*g+1], qy = quat[4*g+2], qz = quat[4*g+3];
    float inv = rsqrtf(qw*qw + qx*qx + qy*qy + qz*qz);
    qw *= inv; qx *= inv; qy *= inv; qz *= inv;

    float r00 = 1.f - 2.f*(qy*qy + qz*qz), r01 = 2.f*(qx*qy - qw*qz), r02 = 2.f*(qx*qz + qw*qy);
    float r10 = 2.f*(qx*qy + qw*qz), r11 = 1.f - 2.f*(qx*qx + qz*qz), r12 = 2.f*(qy*qz - qw*qx);
    float r20 = 2.f*(qx*qz - qw*qy), r21 = 2.f*(qy*qz + qw*qx), r22 = 1.f - 2.f*(qx*qx + qy*qy);

    float sx = scl[3*g+0], sy = scl[3*g+1], sz = scl[3*g+2];
    float s2x = sx*sx, s2y = sy*sy, s2z = sz*sz;
    float px = pos[3*g+0], py = pos[3*g+1], pz = pos[3*g+2];

    const float kchi = 2.7954835f;  // sqrt(7.8147279)
    float rad0 = kchi * sqrtf(r00*r00*s2x + r01*r01*s2y + r02*r02*s2z);
    float rad1 = kchi * sqrtf(r10*r10*s2x + r11*r11*s2y + r12*r12*s2z);
    float rad2 = kchi * sqrtf(r20*r20*s2x + r21*r21*s2y + r22*r22*s2z);

    minmax[0*G+g] = px - rad0;  minmax[1*G+g] = py - rad1;  minmax[2*G+g] = pz - rad2;
    minmax[3*G+g] = px + rad0;  minmax[4*G+g] = py + rad1;  minmax[5*G+g] = pz + rad2;
    minmax[6*G+g] = rad0;

    float ix = 1.f/s2x, iy = 1.f/s2y, iz = 1.f/s2z;
    float C00 = r00*r00*ix + r01*r01*iy + r02*r02*iz;
    float C11 = r10*r10*ix + r11*r11*iy + r12*r12*iz;
    float C22 = r20*r20*ix + r21*r21*iy + r22*r22*iz;
    float C01 = r00*r10*ix + r01*r11*iy + r02*r12*iz;
    float C02 = r00*r20*ix + r01*r21*iy + r02*r22*iz;
    float C12 = r10*r20*ix + r11*r21*iy + r12*r22*iz;

    float bx = C00*px + C01*py + C02*pz;
    float by = C01*px + C11*py + C12*pz;
    float bz = C02*px + C12*py + C22*pz;
    float c0 = bx*px + by*py + bz*pz;
    float logn = -1.5f*LOG_2PI_F - (logf(sx) + logf(sy) + logf(sz));

    psi[ 0*G+g] = -0.5f*C00;  psi[ 1*G+g] = -0.5f*C11;  psi[ 2*G+g] = -0.5f*C22;
    psi[ 3*G+g] = -C01;       psi[ 4*G+g] = -C02;       psi[ 5*G+g] = -C12;
    psi[ 6*G+g] = bx;         psi[ 7*G+g] = by;         psi[ 8*G+g] = bz;
    psi[ 9*G+g] = logn - 0.5f*c0;
    psi[10*G+g] = 0.f;        psi[11*G+g] = 0.f;
}

// ---------------------------------------------------------------------------
// Kernel 2: 7 blocks, each bitonic-sorts one G-length array in LDS and emits
// one order statistic (jnp.quantile 'linear' interpolation).
//   blocks 0..2 -> q=0.01 of min_c axes, 3..5 -> q=0.99 of max_c axes,
//   block 6     -> q=0.50 of radius.x (median)
// ---------------------------------------------------------------------------
__global__ void sort_quantile_kernel(const float* __restrict__ arrays, int G,
                                     float* __restrict__ stats)
{
    __shared__ float buf[2048];
    const float* a = arrays + (size_t)blockIdx.x * G;
    for (int i = threadIdx.x; i < 2048; i += blockDim.x)
        buf[i] = (i < G) ? a[i] : INFINITY;
    __syncthreads();

    for (int k = 2; k <= 2048; k <<= 1) {
        for (int j = k >> 1; j > 0; j >>= 1) {
            for (int i = threadIdx.x; i < 2048; i += blockDim.x) {
                int ixj = i ^ j;
                if (ixj > i) {
                    float x = buf[i], y = buf[ixj];
                    bool up = ((i & k) == 0);
                    if (up ? (x > y) : (x < y)) { buf[i] = y; buf[ixj] = x; }
                }
            }
            __syncthreads();
        }
    }
    if (threadIdx.x == 0) {
        float q = (blockIdx.x < 3) ? 0.01f : ((blockIdx.x < 6) ? 0.99f : 0.5f);
        float posn = q * (float)(G - 1);
        int   i0 = (int)floorf(posn);
        int   i1 = (i0 + 1 < G) ? i0 + 1 : G - 1;
        float t  = posn - (float)i0;
        stats[blockIdx.x] = buf[i0] + t * (buf[i1] - buf[i0]);
    }
}

// ---------------------------------------------------------------------------
// Kernel 3: per-Gaussian voxel bounds + oversized flag.
//   bounds[8g..] = {gx0,gy0,gz0, oversized, gx1,gy1,gz1, 0}
// ---------------------------------------------------------------------------
__global__ void grid_bounds_kernel(const float* __restrict__ minmax,
                                   const float* __restrict__ stats,
                                   int* __restrict__ bounds, int G)
{
    int g = blockIdx.x * blockDim.x + threadIdx.x;
    if (g >= G) return;

    float vox = stats[6] * 3.0f;
    long long nv = 1;
    int lo3[3], hi3[3];
    #pragma unroll
    for (int ax = 0; ax < 3; ++ax) {
        float lo = stats[ax], hi = stats[3+ax];
        float gmn = lo - 0.1f * (hi - lo);
        float mn = minmax[ax*G + g];
        float mx = minmax[(3+ax)*G + g];
        int a0 = (int)floorf((mn - gmn) / vox);
        int a1 = (int)floorf((mx - gmn) / vox);
        a0 = min(max(a0, 0), 1023);
        a1 = min(max(a1, 0), 1023);
        lo3[ax] = a0; hi3[ax] = a1;
        nv *= (long long)(a1 - a0 + 1);
    }
    int over = (nv > 64) ? 1 : 0;
    bounds[8*g+0] = lo3[0]; bounds[8*g+1] = lo3[1]; bounds[8*g+2] = lo3[2];
    bounds[8*g+3] = over;
    bounds[8*g+4] = hi3[0]; bounds[8*g+5] = hi3[1]; bounds[8*g+6] = hi3[2];
    bounds[8*g+7] = 0;
}

// ---------------------------------------------------------------------------
// Kernel 4: scoring GEMM (v_wmma_f32_16x16x4_f32 x3 per tile) + masked top-8.
// One wave per 16 points; loop over G in tiles of 16 gaussians.
// A fragment (16x4 f32): lane m=l&15; VGPR0/1 hold K = kb, kb+1 (kb=2*(l>=16))
// B fragment (4x16 f32): lane n=l&15; same K split.
// D (16x16 f32, 8 VGPRs): lane n=l&15; row r + 8*(l>=16).
// ---------------------------------------------------------------------------
#define WPB 4
__global__ void __launch_bounds__(32 * WPB)
score_topk_kernel(const float* __restrict__ points,
                  const float* __restrict__ psi,
                  const int*   __restrict__ bounds,
                  const float* __restrict__ stats,
                  int G, int P, int K,
                  int* __restrict__ outIds,
                  float* __restrict__ outProbs)
{
    __shared__ float sPhi[WPB][16][12];
    __shared__ float sSc [WPB][16][16];   // [point m][gauss n]
    __shared__ float sMv [WPB][16][16];
    __shared__ int   sMi [WPB][16][16];

    const int tid  = threadIdx.x;
    const int w    = tid >> 5;
    const int l    = tid & 31;
    const int half = l >> 4;
    const int p    = l & 15;              // point row for A/merge, gauss col for B/D
    const int pIdx = (blockIdx.x * WPB + w) * 16 + p;

    // grid geometry (uniform scalars derived from stats)
    float lox = stats[0], loy = stats[1], loz = stats[2];
    float hix = stats[3], hiy = stats[4], hiz = stats[5];
    float vox = stats[6] * 3.0f;
    float gmx = lox - 0.1f * (hix - lox);
    float gmy = loy - 0.1f * (hiy - loy);
    float gmz = loz - 0.1f * (hiz - loz);

    float px = points[3*pIdx+0], py = points[3*pIdx+1], pz = points[3*pIdx+2];
    int pvx = (int)floorf((px - gmx) / vox);
    int pvy = (int)floorf((py - gmy) / vox);
    int pvz = (int)floorf((pz - gmz) / vox);

    if (half == 0) {
        sPhi[w][p][0] = px*px; sPhi[w][p][1] = py*py; sPhi[w][p][2]  = pz*pz;
        sPhi[w][p][3] = px*py; sPhi[w][p][4] = px*pz; sPhi[w][p][5]  = py*pz;
        sPhi[w][p][6] = px;    sPhi[w][p][7] = py;    sPhi[w][p][8]  = pz;
        sPhi[w][p][9] = 1.f;   sPhi[w][p][10] = 0.f;  sPhi[w][p][11] = 0.f;
    }
    __syncthreads();

    const int kb = half * 2;
    v2f a0, a1, a2;
    a0.x = sPhi[w][p][0+kb]; a0.y = sPhi[w][p][1+kb];
    a1.x = sPhi[w][p][4+kb]; a1.y = sPhi[w][p][5+kb];
    a2.x = sPhi[w][p][8+kb]; a2.y = sPhi[w][p][9+kb];

    float tv[8]; int ti[8];
    #pragma unroll
    for (int j = 0; j < 8; ++j) { tv[j] = -INFINITY; ti[j] = -1; }

    for (int g0 = 0; g0 < G; g0 += 16) {
        const float* pc = psi + g0 + p;   // column n = l&15 of Psi
        v2f b0, b1, b2;
        b0.x = pc[(size_t)(0+kb)*G]; b0.y = pc[(size_t)(1+kb)*G];
        b1.x = pc[(size_t)(4+kb)*G]; b1.y = pc[(size_t)(5+kb)*G];
        b2.x = pc[(size_t)(8+kb)*G]; b2.y = pc[(size_t)(9+kb)*G];

        v8f acc = {};
        acc = __builtin_amdgcn_wmma_f32_16x16x4_f32(false, a0, false, b0,
                                                    (short)0, acc, false, false);
        acc = __builtin_amdgcn_wmma_f32_16x16x4_f32(false, a1, false, b1,
                                                    (short)0, acc, false, false);
        acc = __builtin_amdgcn_wmma_f32_16x16x4_f32(false, a2, false, b2,
                                                    (short)0, acc, false, false);

        #pragma unroll
        for (int r = 0; r < 8; ++r) sSc[w][r + 8*half][p] = acc[r];
        __syncthreads();

        // each lane: point p, candidates n = half*8 .. half*8+7
        #pragma unroll
        for (int j = 0; j < 8; ++j) {
            int n = half * 8 + j;
            int g = g0 + n;
            const int* bb = bounds + 8*g;
            bool inb = (pvx >= bb[0]) & (pvx <= bb[4]) &
                       (pvy >= bb[1]) & (pvy <= bb[5]) &
                       (pvz >= bb[2]) & (pvz <= bb[6]);
            float v = (bb[3] | (int)inb) ? sSc[w][p][n] : -INFINITY;
            if (v > tv[7]) {                       // insert into sorted top-8
                float cv = v; int ci = g;
                #pragma unroll
                for (int s = 0; s < 8; ++s) {
                    if (cv > tv[s]) {
                        float tf = tv[s]; tv[s] = cv; cv = tf;
                        int   tn = ti[s]; ti[s] = ci; ci = tn;
                    }
                }
            }
        }
        __syncthreads();
    }

    // merge the two half-lane lists per point
    #pragma unroll
    for (int j = 0; j < 8; ++j) { sMv[w][p][half*8+j] = tv[j]; sMi[w][p][half*8+j] = ti[j]; }
    __syncthreads();

    if (half == 0) {
        int ia = 0, ib = 0;
        for (int k2 = 0; k2 < 8; ++k2) {
            float va = (ia < 8) ? sMv[w][p][ia]     : -INFINITY;
            float vb = (ib < 8) ? sMv[w][p][8 + ib] : -INFINITY;
            float v; int id;
            if (va >= vb) { v = va; id = sMi[w][p][ia];     ++ia; }
            else          { v = vb; id = sMi[w][p][8 + ib]; ++ib; }
            bool fin = isfinite(v);
            if (k2 < K) {
                outIds [(size_t)pIdx * K + k2] = fin ? id : -1;
                outProbs[(size_t)pIdx * K + k2] = fin ? expf(v) : 0.f;
            }
        }
    }
}

// ---------------------------------------------------------------------------
// Host launcher.  Inputs: positions(G,3) scales(G,3) quaternions(G,4)
// points(P,3) K(scalar).  Output: topk_ids(P,K) int32 ++ probs(P,K) f32.
// Workspace: psi 12G f | minmax 7G f | stats 8 f | bounds 8G i  (~220 KB)
// ---------------------------------------------------------------------------
extern "C" void kernel_launch(void* const* d_in, const int* in_sizes, int n_in,
                              void* d_out, int out_size, void* d_ws, size_t ws_size,
                              hipStream_t stream)
{
    const float* positions = (const float*)d_in[0];
    const float* scales    = (const float*)d_in[1];
    const float* quats     = (const float*)d_in[2];
    const float* points    = (const float*)d_in[3];

    int G = in_sizes[0] / 3;
    int P = in_sizes[3] / 3;
    int K = (P > 0) ? (out_size / (2 * P)) : 8;

    float* psi    = (float*)d_ws;
    float* minmax = psi + (size_t)12 * G;
    float* stats  = minmax + (size_t)7 * G;
    int*   bounds = (int*)(stats + 8);

    int*   outIds   = (int*)d_out;
    float* outProbs = (float*)d_out + (size_t)P * K;

    int gb = (G + 255) / 256;
    gauss_setup_kernel<<<gb, 256, 0, stream>>>(positions, scales, quats, psi, minmax, G);
    sort_quantile_kernel<<<7, 256, 0, stream>>>(minmax, G, stats);
    grid_bounds_kernel<<<gb, 256, 0, stream>>>(minmax, stats, bounds, G);

    int nb = (P + 16 * WPB - 1) / (16 * WPB);
    score_topk_kernel<<<nb, 32 * WPB, 0, stream>>>(points, psi, bounds, stats,
                                                   G, P, K, outIds, outProbs);
}